// NCEAverage_21844203668344
// MI455X (gfx1250) — compile-verified
//
#include <hip/hip_runtime.h>
#include <hip/hip_bf16.h>

// Problem constants from the reference
#define NCE_B 256
#define NCE_D 128
#define NCE_KP1 4097            // K+1
#define NCE_TILES 257           // ceil(4097 / 16)
#define NCE_OUTSZ 1281167
#define NCE_INV_T (1.0f / 0.07f)

typedef __attribute__((ext_vector_type(2))) float v2f;
typedef __attribute__((ext_vector_type(8))) float v8f;

__global__ void nce_zero_sum(float* sum_slot) {
    if (threadIdx.x == 0 && blockIdx.x == 0) *sum_slot = 0.0f;
}

// One wave32 handles one (sample b, tile of 16 k-values).
// A (16x4 f32): 16 gathered memory rows, 4-wide K chunk.
// B (4x16 f32): x[b] chunk replicated across all 16 N columns.
// C (16x16 f32): column-replicated dot products, accumulated over D/4 = 32 chunks.
__global__ __launch_bounds__(256) void nce_pass1(
        const float* __restrict__ x,      // (B, D)
        const int*   __restrict__ y,      // (B,)
        const int*   __restrict__ idx,    // (B, K+1)
        const float* __restrict__ memory, // (OUTSZ, D)
        float*       __restrict__ e_out,  // (B, K+1) -> d_out
        float*       __restrict__ sum_slot) {
    const int lane   = threadIdx.x & 31;
    const int wave   = threadIdx.x >> 5;
    const int gwave  = blockIdx.x * (blockDim.x >> 5) + wave;   // 0 .. B*TILES-1
    const int b      = gwave / NCE_TILES;
    const int tile   = gwave - b * NCE_TILES;

    // Row owned by this lane for the A matrix (lanes 0-15 and 16-31 both map M = lane&15)
    const int m      = lane & 15;
    int k            = tile * 16 + m;
    const bool krow_valid = (k < NCE_KP1);
    int kc           = krow_valid ? k : 0;                      // clamp tail to a safe gather
    int row          = (kc == 0) ? y[b] : idx[b * NCE_KP1 + kc];

    const float* __restrict__ rowp = memory + (size_t)row * NCE_D;
    const float* __restrict__ xb   = x + (size_t)b * NCE_D;
    const int off = (lane < 16) ? 0 : 2;   // K slots 0,1 on lanes 0-15; 2,3 on lanes 16-31

    v8f acc = {};
    #pragma unroll
    for (int c4 = 0; c4 < NCE_D / 4; ++c4) {
        const v2f a  = *(const v2f*)(rowp + c4 * 4 + off);   // A[m][c4*4+off .. +1]
        const v2f bv = *(const v2f*)(xb   + c4 * 4 + off);   // B[k][*] = x[b][k] (all N cols)
        acc = __builtin_amdgcn_wmma_f32_16x16x4_f32(
            /*neg_a=*/false, a, /*neg_b=*/false, bv,
            /*c_mod=*/(short)0, acc, /*reuse_a=*/false, /*reuse_b=*/false);
    }

    // C layout: lanes 0-15 hold N=lane, VGPR r = M=r; lanes 16-31 hold N=lane-16, VGPR r = M=8+r.
    // All N columns are identical, so lanes 0..7 emit rows 0..7 and lanes 16..23 emit rows 8..15.
    const int r = lane & 7;
    float v = acc[0];
    #pragma unroll
    for (int i = 1; i < 8; ++i) v = (r == i) ? acc[i] : v;

    const bool producer = (lane < 8) || (lane >= 16 && lane < 24);
    const int  rowm     = (lane < 16) ? r : (8 + r);
    const int  k_out    = tile * 16 + rowm;
    const bool out_ok   = producer && (k_out < NCE_KP1);

    float e = out_ok ? __expf(v * NCE_INV_T) : 0.0f;
    if (out_ok) e_out[(size_t)b * NCE_KP1 + k_out] = e;

    // Block reduction of sum(e), one global atomic per block.
    __shared__ float red[256];
    red[threadIdx.x] = e;
    __syncthreads();
    for (int s = 128; s > 0; s >>= 1) {
        if ((int)threadIdx.x < s) red[threadIdx.x] += red[threadIdx.x + s];
        __syncthreads();
    }
    if (threadIdx.x == 0) atomicAdd(sum_slot, red[0]);
}

__global__ void nce_scale(float* __restrict__ out, const float* __restrict__ sum_slot) {
    const int i = blockIdx.x * blockDim.x + threadIdx.x;
    const int n = NCE_B * NCE_KP1;
    if (i < n) {
        const float s = *sum_slot;
        // out = e / Z,  Z = (sum / (B*(K+1))) * outputSize
        const float scale = (float)n / (s * (float)NCE_OUTSZ);
        out[i] *= scale;
    }
}

extern "C" void kernel_launch(void* const* d_in, const int* in_sizes, int n_in,
                              void* d_out, int out_size, void* d_ws, size_t ws_size,
                              hipStream_t stream) {
    const float* x      = (const float*)d_in[0];
    const int*   y      = (const int*)d_in[1];
    const int*   idx    = (const int*)d_in[2];
    const float* memory = (const float*)d_in[3];
    float*       out    = (float*)d_out;
    float*       sum    = (float*)d_ws;      // one float of scratch

    nce_zero_sum<<<1, 32, 0, stream>>>(sum);

    const int total_waves  = NCE_B * NCE_TILES;      // 65792
    const int waves_per_bl = 256 / 32;               // 8
    const int grid1        = total_waves / waves_per_bl; // 8224 (exact)
    nce_pass1<<<grid1, 256, 0, stream>>>(x, y, idx, memory, out, sum);

    const int n     = NCE_B * NCE_KP1;               // 1,048,832
    const int grid2 = (n + 255) / 256;
    nce_scale<<<grid2, 256, 0, stream>>>(out, sum);
}